// SpectralEdgeGNN_81544249082549
// MI455X (gfx1250) — compile-verified
//
#include <hip/hip_runtime.h>
#include <hip/hip_bf16.h>
#include <math.h>

typedef __attribute__((ext_vector_type(2))) float v2f;
typedef __attribute__((ext_vector_type(8))) float v8f;

#define F_IN 128
#define HID  256
#define HID2 128
#define NCLS 40
#define NCLSP 48   // padded to 3 column tiles

// ---------------- utility ----------------
__global__ void zero_kernel(float* __restrict__ p, long long count) {
    long long i = blockIdx.x * (long long)blockDim.x + threadIdx.x;
    if (i < count) p[i] = 0.0f;
}

__global__ void pad_wc_kernel(const float* __restrict__ Wc, const float* __restrict__ bc,
                              float* __restrict__ Wp, float* __restrict__ bp,
                              int K, int C, int Cp) {
    int i = blockIdx.x * blockDim.x + threadIdx.x;
    int tot = K * Cp;
    if (i < tot) {
        int r = i / Cp, c = i - r * Cp;
        Wp[i] = (c < C) ? Wc[r * C + c] : 0.0f;
    }
    if (i < Cp) bp[i] = (i < C) ? bc[i] : 0.0f;
}

// ---------------- graph normalization ----------------
__global__ void deg_kernel(const int* __restrict__ src, const float* __restrict__ ew,
                           float* __restrict__ deg, int E) {
    int e = blockIdx.x * blockDim.x + threadIdx.x;
    if (e < E) atomicAdd(&deg[src[e]], ew[e]);
}

__global__ void dinv_kernel(float* __restrict__ deg, int n) {
    int i = blockIdx.x * blockDim.x + threadIdx.x;
    if (i < n) {
        float d = deg[i];
        deg[i] = (d > 0.0f) ? rsqrtf(d) : 0.0f;   // in-place: deg becomes dinv
    }
}

__global__ void norm_kernel(const int* __restrict__ src, const int* __restrict__ dst,
                            const float* __restrict__ ew, const float* __restrict__ dinv,
                            float* __restrict__ norm, int E) {
    int e = blockIdx.x * blockDim.x + threadIdx.x;
    if (e < E) norm[e] = -dinv[src[e]] * ew[e] * dinv[dst[e]];
}

// ---------------- edge scatter: out[dst] += norm * x[src] ----------------
__global__ void scatter_kernel(const float* __restrict__ x, const int* __restrict__ src,
                               const int* __restrict__ dst, const float* __restrict__ norm,
                               float* __restrict__ out, int E, int F) {
    long long t = blockIdx.x * (long long)blockDim.x + threadIdx.x;
    int chunks = F >> 2;
    long long e = t / chunks;
    if (e >= E) return;
    int ch = (int)(t - e * chunks) << 2;
    float w = norm[e];
    int s = src[e], d = dst[e];
    const float4 v = *reinterpret_cast<const float4*>(x + (long long)s * F + ch);
    float* o = out + (long long)d * F + ch;
    atomicAdd(o + 0, w * v.x);
    atomicAdd(o + 1, w * v.y);
    atomicAdd(o + 2, w * v.z);
    atomicAdd(o + 3, w * v.w);
}

// ---------------- fp32 WMMA GEMM: out = X*W0 (+ TX*W1) + bias ----------------
// One wave computes NT adjacent 16x16 output tiles (shared A fragment).
// V_WMMA_F32_16X16X4_F32 layouts:
//   A (16x4): lanes 0-15 M=lane, v0=K0,v1=K1; lanes 16-31 K2,K3.
//   B (4x16): v0 lanes 0-15 = row K0, lanes 16-31 = row K1; v1: K2/K3.
//   C/D: VGPR v -> M=v (lanes 0-15) / M=v+8 (lanes 16-31), N=lane&15.
// Fout   = padded column count / W row stride (multiple of 16, loads unguarded)
// FoutR  = real column count (store/bias guard)
// outStr = output row stride
template <int NT, bool HAS_TX>
__global__ void gemm_cheb_kernel(const float* __restrict__ X,
                                 const float* __restrict__ TX,
                                 const float* __restrict__ W0,
                                 const float* __restrict__ W1,
                                 const float* __restrict__ bias,
                                 float* __restrict__ out,
                                 int n, int K, int Fout, int FoutR, int outStr) {
    const int lane  = threadIdx.x & 31;
    const int wave  = threadIdx.x >> 5;
    const int rowTile = blockIdx.x;
    const int wavesPerBlock = blockDim.x >> 5;
    const int tile0 = (blockIdx.y * wavesPerBlock + wave) * NT;
    if (tile0 * 16 >= Fout) return;                   // wave-uniform exit

    const int m     = lane & 15;
    const int khalf = lane >> 4;                      // 0 or 1
    int row = rowTile * 16 + m;
    if (row >= n) row = n - 1;                        // keep EXEC full for WMMA
    const int col0 = tile0 * 16 + m;

    const float* __restrict__ xr  = X + (long long)row * K + khalf * 2;
    const float* __restrict__ txr = HAS_TX ? TX + (long long)row * K + khalf * 2 : nullptr;
    const float* __restrict__ w0p = W0 + (long long)khalf * Fout + col0;
    const float* __restrict__ w1p = HAS_TX ? W1 + (long long)khalf * Fout + col0 : nullptr;

    v8f acc[NT];
    #pragma unroll
    for (int t = 0; t < NT; ++t) acc[t] = (v8f){};

    #pragma unroll 4
    for (int kb = 0; kb < K; kb += 4) {
        v2f a;
        a.x = xr[kb + 0];
        a.y = xr[kb + 1];
        v2f b[NT];
        #pragma unroll
        for (int t = 0; t < NT; ++t) {
            b[t].x = w0p[(long long)kb * Fout + t * 16];
            b[t].y = w0p[(long long)(kb + 2) * Fout + t * 16];
        }
        #pragma unroll
        for (int t = 0; t < NT; ++t)
            acc[t] = __builtin_amdgcn_wmma_f32_16x16x4_f32(false, a, false, b[t],
                                                           (short)0, acc[t], false, false);
        if (HAS_TX) {
            v2f a2;
            a2.x = txr[kb + 0];
            a2.y = txr[kb + 1];
            v2f b2[NT];
            #pragma unroll
            for (int t = 0; t < NT; ++t) {
                b2[t].x = w1p[(long long)kb * Fout + t * 16];
                b2[t].y = w1p[(long long)(kb + 2) * Fout + t * 16];
            }
            #pragma unroll
            for (int t = 0; t < NT; ++t)
                acc[t] = __builtin_amdgcn_wmma_f32_16x16x4_f32(false, a2, false, b2[t],
                                                               (short)0, acc[t], false, false);
        }
    }

    #pragma unroll
    for (int t = 0; t < NT; ++t) {
        const int col = col0 + t * 16;
        if (col < FoutR) {
            const float bb = bias[col];
            #pragma unroll
            for (int v = 0; v < 8; ++v) {
                int r = rowTile * 16 + v + khalf * 8;
                if (r < n) out[(long long)r * outStr + col] = acc[t][v] + bb;
            }
        }
    }
}

// ---------------- BatchNorm statistics (sum, sumsq per channel) ----------------
__global__ void bn_stats_kernel(const float* __restrict__ h, float* __restrict__ sum,
                                float* __restrict__ sumsq, int n, int F, int rowsPerBlock) {
    const int tid = threadIdx.x;
    const int c = tid % F;
    const int rOff = tid / F;
    const int rStep = blockDim.x / F;
    const int r0 = blockIdx.x * rowsPerBlock;
    int r1 = r0 + rowsPerBlock; if (r1 > n) r1 = n;
    float s = 0.0f, ss = 0.0f;
    for (int r = r0 + rOff; r < r1; r += rStep) {
        float v = h[(long long)r * F + c];
        s += v; ss += v * v;
    }
    atomicAdd(&sum[c], s);
    atomicAdd(&sumsq[c], ss);
}

__global__ void bn_apply_relu_kernel(float* __restrict__ h, const float* __restrict__ sum,
                                     const float* __restrict__ sumsq,
                                     const float* __restrict__ g, const float* __restrict__ beta,
                                     int n, int F) {
    long long idx = blockIdx.x * (long long)blockDim.x + threadIdx.x;
    long long tot = (long long)n * F;
    if (idx >= tot) return;
    int c = (int)(idx % F);
    float inv_n = 1.0f / (float)n;
    float mu = sum[c] * inv_n;
    float var = sumsq[c] * inv_n - mu * mu;
    var = var > 0.0f ? var : 0.0f;
    float y = g[c] * (h[idx] - mu) * rsqrtf(var + 1e-5f) + beta[c];
    h[idx] = y > 0.0f ? y : 0.0f;
}

// ---------------- log-softmax over 40 classes ----------------
__global__ void logsoftmax_kernel(const float* __restrict__ logits, float* __restrict__ out,
                                  int n, int C) {
    int r = blockIdx.x * blockDim.x + threadIdx.x;
    if (r >= n) return;
    const float* p = logits + (long long)r * C;
    float m = -3.402823466e38f;
    for (int c = 0; c < C; ++c) m = fmaxf(m, p[c]);
    float s = 0.0f;
    for (int c = 0; c < C; ++c) s += expf(p[c] - m);
    float lse = m + logf(s);
    float* o = out + (long long)r * C;
    for (int c = 0; c < C; ++c) o[c] = p[c] - lse;
}

// ---------------- launch ----------------
static inline unsigned cdiv(long long a, long long b) { return (unsigned)((a + b - 1) / b); }

extern "C" void kernel_launch(void* const* d_in, const int* in_sizes, int n_in,
                              void* d_out, int out_size, void* d_ws, size_t ws_size,
                              hipStream_t stream) {
    const float* x   = (const float*)d_in[0];
    const int*   ei  = (const int*)  d_in[1];
    const float* ew  = (const float*)d_in[2];
    const float* W1  = (const float*)d_in[3];   // (2,128,256)
    const float* b1  = (const float*)d_in[4];
    const float* g1  = (const float*)d_in[5];
    const float* bt1 = (const float*)d_in[6];
    const float* W2  = (const float*)d_in[7];   // (2,256,128)
    const float* b2  = (const float*)d_in[8];
    const float* g2  = (const float*)d_in[9];
    const float* bt2 = (const float*)d_in[10];
    const float* Wc  = (const float*)d_in[11];  // (128,40)
    const float* bc  = (const float*)d_in[12];
    float* out = (float*)d_out;

    const int n = in_sizes[0] / F_IN;
    const int E = in_sizes[1] / 2;
    const int* srcI = ei;
    const int* dstI = ei + E;

    float* ws    = (float*)d_ws;
    float* deg   = ws;                         // n (becomes dinv)
    float* norm  = deg + n;                    // E
    float* sum   = norm + E;                   // 256
    float* sumsq = sum + 256;                  // 256
    float* Wp    = sumsq + 256;                // 128*48 padded classifier weights
    float* bp    = Wp + F_IN * NCLSP;          // 48
    float* bufA  = bp + NCLSP;                 // n*256  (tx1 / tx2 / logits)
    float* bufB  = bufA + (size_t)n * 256;     // n*256  (h1)
    float* bufC  = bufB + (size_t)n * 256;     // n*128  (h2)

    const int B = 256;
    const int GB = 128;                        // 4 waves per GEMM block
    const int rowTiles = cdiv(n, 16);

    // --- normalization coefficients + padded classifier weights ---
    zero_kernel<<<cdiv(n, B), B, 0, stream>>>(deg, n);
    deg_kernel<<<cdiv(E, B), B, 0, stream>>>(srcI, ew, deg, E);
    dinv_kernel<<<cdiv(n, B), B, 0, stream>>>(deg, n);
    norm_kernel<<<cdiv(E, B), B, 0, stream>>>(srcI, dstI, ew, deg, norm, E);
    pad_wc_kernel<<<cdiv(F_IN * NCLSP, B), B, 0, stream>>>(Wc, bc, Wp, bp, F_IN, NCLS, NCLSP);

    // --- layer 1: tx1 = L_hat x ; h1 = x*W1[0] + tx1*W1[1] + b1 ; BN+ReLU ---
    zero_kernel<<<cdiv((long long)n * F_IN, B), B, 0, stream>>>(bufA, (long long)n * F_IN);
    scatter_kernel<<<cdiv((long long)E * (F_IN / 4), B), B, 0, stream>>>(
        x, srcI, dstI, norm, bufA, E, F_IN);
    gemm_cheb_kernel<4, true><<<dim3(rowTiles, 1), GB, 0, stream>>>(
        x, bufA, W1, W1 + (size_t)F_IN * HID, b1, bufB, n, F_IN, HID, HID, HID);
    zero_kernel<<<2, B, 0, stream>>>(sum, 512);
    bn_stats_kernel<<<cdiv(n, 512), B, 0, stream>>>(bufB, sum, sumsq, n, HID, 512);
    bn_apply_relu_kernel<<<cdiv((long long)n * HID, B), B, 0, stream>>>(
        bufB, sum, sumsq, g1, bt1, n, HID);

    // --- layer 2: tx2 = L_hat h1 ; h2 = h1*W2[0] + tx2*W2[1] + b2 ; BN+ReLU ---
    zero_kernel<<<cdiv((long long)n * HID, B), B, 0, stream>>>(bufA, (long long)n * HID);
    scatter_kernel<<<cdiv((long long)E * (HID / 4), B), B, 0, stream>>>(
        bufB, srcI, dstI, norm, bufA, E, HID);
    gemm_cheb_kernel<2, true><<<dim3(rowTiles, 1), GB, 0, stream>>>(
        bufB, bufA, W2, W2 + (size_t)HID * HID2, b2, bufC, n, HID, HID2, HID2, HID2);
    zero_kernel<<<2, B, 0, stream>>>(sum, 512);
    bn_stats_kernel<<<cdiv(n, 512), B, 0, stream>>>(bufC, sum, sumsq, n, HID2, 512);
    bn_apply_relu_kernel<<<cdiv((long long)n * HID2, B), B, 0, stream>>>(
        bufC, sum, sumsq, g2, bt2, n, HID2);

    // --- classifier + log_softmax ---
    gemm_cheb_kernel<1, false><<<dim3(rowTiles, 1), GB, 0, stream>>>(
        bufC, nullptr, Wp, nullptr, bp, bufA /*logits*/, n, HID2, NCLSP, NCLS, NCLS);
    logsoftmax_kernel<<<cdiv(n, B), B, 0, stream>>>(bufA, out, n, NCLS);
}